// criterion_mat_65695819760238
// MI455X (gfx1250) — compile-verified
//
#include <hip/hip_runtime.h>
#include <hip/hip_bf16.h>
#include <stdint.h>

#define NCLS 1000
#define NDF  256
#define BSZ  512
#define ALPC 0.1f

typedef __attribute__((ext_vector_type(2))) float v2f;
typedef __attribute__((ext_vector_type(8))) float v8f;

// ---------------------------------------------------------------------------
// Kernel 1: sequential per-class statistics (single wave32).
// Produces: Amat[i][d] = f_i[d] - new_mean[d]  (0 for first occurrence),
//           occ[i] = occurrence index (1-based) of sample i within its class,
//           prev[i] = previous sample index of the same class (-1 if none).
// mean lives in global ws; each feature dim d is owned by lane d%32, so all
// mean accesses are per-thread same-address (in-order). counts/last live in
// LDS (in-order per wave).
// ---------------------------------------------------------------------------
__global__ __launch_bounds__(32) void stats_kernel(
    const float* __restrict__ df, const int* __restrict__ gt,
    float* __restrict__ mean, float* __restrict__ Amat,
    int* __restrict__ occ, int* __restrict__ prev)
{
    __shared__ int counts[NCLS];
    __shared__ int last[NCLS];
    const int lane = threadIdx.x;

    for (int t = lane; t < NCLS; t += 32) { counts[t] = 0; last[t] = -1; }

    for (int i = 0; i < BSZ; ++i) {
        const int t  = gt[i];
        const int it = counts[t];                       // LDS, in-order per wave
        const float c1 = (float)it / (float)(it + 1);
        const float c2 = 1.0f / (float)(it + 1);

        #pragma unroll
        for (int r = 0; r < NDF / 32; ++r) {
            const int d = lane + r * 32;
            const float f  = df[i * NDF + d];
            const float nm = (it == 0) ? f : (mean[t * NDF + d] * c1 + f * c2);
            mean[t * NDF + d] = nm;
            Amat[i * NDF + d] = (it == 0) ? 0.0f : (f - nm);
        }
        // all lanes write identical scalar values (benign, keeps per-lane
        // store->load ordering trivially correct)
        occ[i]    = it + 1;
        prev[i]   = last[t];
        counts[t] = it + 1;
        last[t]   = i;
    }
}

// ---------------------------------------------------------------------------
// Kernel 2: U = FC[1000,256] @ X^T, X in {df, Amat} (both [512,256] row-major,
// which is exactly B^T for the WMMA). One wave per 16x16 output tile using
// V_WMMA_F32_16X16X4_F32 (full f32 precision, 64 MACs per tile).
//   A frag: lane l -> float2 fc[row0 + l%16][kb + 2*(l/16) ..+1]
//   B frag: lane l -> float2 X [col0 + l%16][kb + 2*(l/16) ..+1]
//   D:      vgpr v -> (m = row0 + v + 8*(l/16), n = col0 + l%16)
// Two independent accumulators over alternating 4-wide K steps break the WMMA->WMMA RAW chain;
// boundary predication only on the single partial M-tile (wave-uniform test).
// ---------------------------------------------------------------------------
__global__ __launch_bounds__(32) void wmma_gemm_kernel(
    const float* __restrict__ fc, const float* __restrict__ df,
    const float* __restrict__ Amat,
    float* __restrict__ Uf, float* __restrict__ Ua)
{
    const int lane = threadIdx.x;
    const int half = lane >> 4;      // 0 or 1
    const int lm   = lane & 15;
    const int col0 = blockIdx.x * 16;
    const int row0 = blockIdx.y * 16;

    const float* __restrict__ X = (blockIdx.z == 0) ? df : Amat;
    float*       __restrict__ U = (blockIdx.z == 0) ? Uf : Ua;

    const int m  = row0 + lm;
    const int mc = (m < NCLS) ? m : (NCLS - 1);          // clamp: rows >=1000 unused
    const float* arow = fc + (size_t)mc * NDF + 2 * half;
    const float* brow = X  + (size_t)(col0 + lm) * NDF + 2 * half;

    v8f acc0 = {};
    v8f acc1 = {};
    #pragma unroll 4
    for (int kb = 0; kb < NDF; kb += 8) {
        v2f a0 = *(const v2f*)(arow + kb);
        v2f b0 = *(const v2f*)(brow + kb);
        v2f a1 = *(const v2f*)(arow + kb + 4);
        v2f b1 = *(const v2f*)(brow + kb + 4);
        acc0 = __builtin_amdgcn_wmma_f32_16x16x4_f32(
            false, a0, false, b0, (short)0, acc0, false, false);
        acc1 = __builtin_amdgcn_wmma_f32_16x16x4_f32(
            false, a1, false, b1, (short)0, acc1, false, false);
    }

    float* urow = U + (size_t)(row0 + 8 * half) * BSZ + col0 + lm;
    if (row0 + 16 <= NCLS) {
        // full tile: unconditional, clause-friendly stores
        #pragma unroll
        for (int v = 0; v < 8; ++v)
            urow[(size_t)v * BSZ] = acc0[v] + acc1[v];
    } else {
        // single partial M-tile at the boundary
        #pragma unroll
        for (int v = 0; v < 8; ++v) {
            const int mr = row0 + v + 8 * half;
            if (mr < NCLS) urow[(size_t)v * BSZ] = acc0[v] + acc1[v];
        }
    }
}

// ---------------------------------------------------------------------------
// Kernel 3: finalize. z_i[c] = Uf[c,i] - Uf[t,i] + 0.5*ALP * quad, where
// quad walks the same-class occurrence chain of sample i:
//   w(kj,ki) = (kj-1)/(kj*ki),  g = Ua[c,j] - Ua[t,j],  quad += w*g*g
// Threads: tid = i*NCLS + c (coalesced output).
// ---------------------------------------------------------------------------
__global__ __launch_bounds__(256) void finalize_kernel(
    const float* __restrict__ Uf, const float* __restrict__ Ua,
    const int* __restrict__ gt, const int* __restrict__ occ,
    const int* __restrict__ prev, float* __restrict__ out)
{
    const int tid = blockIdx.x * blockDim.x + threadIdx.x;
    if (tid >= BSZ * NCLS) return;
    const int i = tid / NCLS;
    const int c = tid - i * NCLS;
    const int t = gt[i];

    float z = Uf[(size_t)c * BSZ + i] - Uf[(size_t)t * BSZ + i];

    const int ki = occ[i];
    if (ki >= 2) {
        const float inv_ki = 1.0f / (float)ki;
        float quad = 0.0f;
        int j = i;
        while (j >= 0) {
            const int kj = occ[j];
            if (kj >= 2) {
                const float g = Ua[(size_t)c * BSZ + j] - Ua[(size_t)t * BSZ + j];
                const float w = ((float)(kj - 1) / (float)kj) * inv_ki;
                quad += w * g * g;
            }
            j = prev[j];
        }
        z += 0.5f * ALPC * quad;
    }
    out[tid] = z;
}

// ---------------------------------------------------------------------------
extern "C" void kernel_launch(void* const* d_in, const int* in_sizes, int n_in,
                              void* d_out, int out_size, void* d_ws, size_t ws_size,
                              hipStream_t stream)
{
    const float* df = (const float*)d_in[0];   // [512, 256]
    const int*   gt = (const int*)  d_in[1];   // [512]
    const float* fc = (const float*)d_in[2];   // [1000, 256]
    float* out = (float*)d_out;                // [512*1000]

    char* ws = (char*)d_ws;
    size_t off = 0;
    auto take = [&](size_t bytes) -> char* {
        char* p = ws + off;
        off = (off + bytes + 255) & ~(size_t)255;
        return p;
    };
    float* mean = (float*)take((size_t)NCLS * NDF * sizeof(float));   // 1.0 MB
    float* Amat = (float*)take((size_t)BSZ  * NDF * sizeof(float));   // 0.5 MB
    float* Uf   = (float*)take((size_t)NCLS * BSZ * sizeof(float));   // 2.0 MB
    float* Ua   = (float*)take((size_t)NCLS * BSZ * sizeof(float));   // 2.0 MB
    int*   occ  = (int*)  take((size_t)BSZ * sizeof(int));
    int*   prev = (int*)  take((size_t)BSZ * sizeof(int));

    hipLaunchKernelGGL(stats_kernel, dim3(1), dim3(32), 0, stream,
                       df, gt, mean, Amat, occ, prev);

    hipLaunchKernelGGL(wmma_gemm_kernel,
                       dim3(BSZ / 16, (NCLS + 15) / 16, 2), dim3(32), 0, stream,
                       fc, df, Amat, Uf, Ua);

    const int total = BSZ * NCLS;
    hipLaunchKernelGGL(finalize_kernel, dim3((total + 255) / 256), dim3(256), 0, stream,
                       Uf, Ua, gt, occ, prev, out);
}